// MixHopConv_3951369912457
// MI455X (gfx1250) — compile-verified
//
#include <hip/hip_runtime.h>

#define N_NODES 100000
#define N_EDGES 1600000
#define DIM 128
#define OUT_STRIDE 384
#define ROW_TILES (N_NODES / 16)   // 6250

typedef float v2f __attribute__((ext_vector_type(2)));
typedef float v8f __attribute__((ext_vector_type(8)));

// ---------------- utility kernels ----------------

__global__ void zero4_kernel(float4* __restrict__ p, int n4) {
    int i = blockIdx.x * blockDim.x + threadIdx.x;
    if (i < n4) p[i] = make_float4(0.f, 0.f, 0.f, 0.f);
}

__global__ void degree_kernel(const int* __restrict__ dst, float* __restrict__ deg, int n) {
    int i = blockIdx.x * blockDim.x + threadIdx.x;
    if (i < n) unsafeAtomicAdd(&deg[dst[i]], 1.0f);
}

__global__ void norm_kernel(const float* __restrict__ deg, float* __restrict__ nrm, int n) {
    int i = blockIdx.x * blockDim.x + threadIdx.x;
    if (i < n) nrm[i] = 1.0f / sqrtf(fmaxf(deg[i], 1.0f));
}

// hout[dst] += hin[src] * nrm[src] ; one wave (32 lanes) per edge, float4 per lane
__global__ void scatter_kernel(const float* __restrict__ hin, const float* __restrict__ nrm,
                               const int* __restrict__ src, const int* __restrict__ dst,
                               float* __restrict__ hout, int nEdges) {
    int e = blockIdx.x * (blockDim.x >> 5) + (threadIdx.x >> 5);
    if (e >= nEdges) return;
    int lane = threadIdx.x & 31;
    int s = src[e];
    int d = dst[e];
    float ns = nrm[s];
    float4 v = reinterpret_cast<const float4*>(hin + (size_t)s * DIM)[lane];
    float* p = hout + (size_t)d * DIM + lane * 4;
    unsafeAtomicAdd(p + 0, v.x * ns);
    unsafeAtomicAdd(p + 1, v.y * ns);
    unsafeAtomicAdd(p + 2, v.z * ns);
    unsafeAtomicAdd(p + 3, v.w * ns);
}

// h[node][:] *= nrm[node], one float4 per thread
__global__ void scale_kernel(float* __restrict__ h, const float* __restrict__ nrm, int n4) {
    int i = blockIdx.x * blockDim.x + threadIdx.x;
    if (i >= n4) return;
    int node = i >> 5;                 // 32 float4 chunks per 128-float row
    float s = nrm[node];
    float4* p = reinterpret_cast<float4*>(h) + i;
    float4 v = *p;
    v.x *= s; v.y *= s; v.z *= s; v.w *= s;
    *p = v;
}

// ---------------- WMMA f32 GEMM ----------------
// out[:, z*128 : z*128+128] = Hz @ Wz, Hz in {feats, h1, h2}
// One wave computes a 16x64 output strip via 4 accumulator tiles of
// V_WMMA_F32_16X16X4_F32. blockDim = 128 (4 waves), each wave = 1 row tile.
__global__ void gemm_wmma_kernel(const float* __restrict__ h0,
                                 const float* __restrict__ h1,
                                 const float* __restrict__ h2,
                                 const float* __restrict__ W0,
                                 const float* __restrict__ W1,
                                 const float* __restrict__ W2,
                                 float* __restrict__ out) {
    const int z = blockIdx.z;
    const float* __restrict__ A = (z == 0) ? h0 : (z == 1) ? h1 : h2;
    const float* __restrict__ W = (z == 0) ? W0 : (z == 1) ? W1 : W2;
    const int colOffset = z * DIM;

    const int lane = threadIdx.x & 31;
    const int wid = threadIdx.x >> 5;
    const int rowTile = blockIdx.x * 4 + wid;
    if (rowTile >= ROW_TILES) return;      // wave-uniform branch, EXEC stays full

    const int half = lane >> 4;            // 0: K pair {0,1}; 1: K pair {2,3}
    const int m = lane & 15;               // A row (in tile) / B,C column
    const int colBase = blockIdx.y * 64;   // this wave: cols colBase..colBase+63
    const int row0 = rowTile * 16;

    v8f acc[4];
#pragma unroll
    for (int t = 0; t < 4; ++t) acc[t] = (v8f){0.f, 0.f, 0.f, 0.f, 0.f, 0.f, 0.f, 0.f};

    const float* __restrict__ arow = A + (size_t)(row0 + m) * DIM;

    for (int k = 0; k < DIM; k += 4) {
        const int ka = k + 2 * half;
        v2f a = *(const v2f*)(arow + ka);              // A[m][ka..ka+1], 8B aligned
        const float* __restrict__ wk = W + (size_t)ka * DIM;
#pragma unroll
        for (int t = 0; t < 4; ++t) {
            const int n = colBase + t * 16 + m;
            v2f b;
            b.x = wk[n];                               // B[ka][n]
            b.y = wk[DIM + n];                         // B[ka+1][n]
            acc[t] = __builtin_amdgcn_wmma_f32_16x16x4_f32(
                false, a, false, b, (short)0, acc[t], false, false);
        }
    }

#pragma unroll
    for (int t = 0; t < 4; ++t) {
#pragma unroll
        for (int v = 0; v < 8; ++v) {
            const int row = row0 + v + 8 * half;
            const int col = colOffset + colBase + t * 16 + m;
            out[(size_t)row * OUT_STRIDE + col] = acc[t][v];
        }
    }
}

// ---------------- launch ----------------

extern "C" void kernel_launch(void* const* d_in, const int* in_sizes, int n_in,
                              void* d_out, int out_size, void* d_ws, size_t ws_size,
                              hipStream_t stream) {
    const float* feats = (const float*)d_in[0];
    const float* W0 = (const float*)d_in[1];
    const float* W1 = (const float*)d_in[2];
    const float* W2 = (const float*)d_in[3];
    const int* src = (const int*)d_in[4];
    const int* dst = (const int*)d_in[5];
    float* out = (float*)d_out;

    float* deg = (float*)d_ws;
    float* nrm = deg + 131072;
    float* h1 = nrm + 131072;
    float* h2 = h1 + (size_t)N_NODES * DIM;

    const int hFloats = N_NODES * DIM;         // 12.8M
    const int h4 = hFloats / 4;                // 3.2M float4
    const int edgeBlocks = (N_EDGES + 7) / 8;  // 8 waves per 256-thread block

    // degree + norm
    zero4_kernel<<<(N_NODES / 4 + 255) / 256, 256, 0, stream>>>((float4*)deg, N_NODES / 4);
    degree_kernel<<<(N_EDGES + 255) / 256, 256, 0, stream>>>(dst, deg, N_EDGES);
    norm_kernel<<<(N_NODES + 255) / 256, 256, 0, stream>>>(deg, nrm, N_NODES);

    // hop 1: h1 = norm * A(norm * feats)
    zero4_kernel<<<(h4 + 255) / 256, 256, 0, stream>>>((float4*)h1, h4);
    scatter_kernel<<<edgeBlocks, 256, 0, stream>>>(feats, nrm, src, dst, h1, N_EDGES);
    scale_kernel<<<(N_NODES * 32 + 255) / 256, 256, 0, stream>>>(h1, nrm, N_NODES * 32);

    // hop 2: h2 = norm * A(norm * h1)
    zero4_kernel<<<(h4 + 255) / 256, 256, 0, stream>>>((float4*)h2, h4);
    scatter_kernel<<<edgeBlocks, 256, 0, stream>>>(h1, nrm, src, dst, h2, N_EDGES);
    scale_kernel<<<(N_NODES * 32 + 255) / 256, 256, 0, stream>>>(h2, nrm, N_NODES * 32);

    // 3 GEMMs fused into one launch: grid.z selects (H, W, column slab)
    dim3 ggrid((ROW_TILES + 3) / 4, 2, 3);
    gemm_wmma_kernel<<<ggrid, 128, 0, stream>>>(feats, h1, h2, W0, W1, W2, out);
}